// Linear_act_sp_24215025615322
// MI455X (gfx1250) — compile-verified
//
#include <hip/hip_runtime.h>

// ---------------------------------------------------------------------------
// Problem constants (reference: x[4,8192,1024] fp32, W[1024,1024] fp32)
// ---------------------------------------------------------------------------
#define DIM   1024          // reduction / feature dim (expanded K)
#define PKD   512           // packed K per row (2:4 -> half)
#define IPR   32            // index u32 words per row (256 groups * 4 bits)
#define NOUT  1024          // output features
#define BM    64            // block tile M
#define BN    128           // block tile N
#define BK    64            // expanded K step (== SWMMAC K)
#define BKP   32            // packed K step
#define NK    (DIM / BK)    // 16 K-steps
#define VAR_EPS 1e-9f

typedef __attribute__((ext_vector_type(32))) _Float16 v32h;
typedef __attribute__((ext_vector_type(16))) _Float16 v16h;
typedef __attribute__((ext_vector_type(8)))  _Float16 v8h;
typedef __attribute__((ext_vector_type(4)))  _Float16 v4h;
typedef __attribute__((ext_vector_type(2)))  _Float16 v2h;
typedef __attribute__((ext_vector_type(8)))  float    v8f;

// ---------------------------------------------------------------------------
// Pass 1: per-row 2:4 magnitude prune + variance-ratio scale.
// Emits packed values f16 [T, 512] (scaled) + 2-bit metadata [T, 32] u32.
// ---------------------------------------------------------------------------
__global__ __launch_bounds__(256)
void prep_kernel(const float* __restrict__ x, _Float16* __restrict__ apack,
                 unsigned int* __restrict__ idxbuf)
{
    __shared__ float red[4][256];
    __shared__ unsigned int codes[256];
    __shared__ float vsh;

    const int t   = blockIdx.x;
    const int tid = threadIdx.x;

    const float4 g = reinterpret_cast<const float4*>(x + (size_t)t * DIM)[tid];

    const float a0 = fabsf(g.x), a1 = fabsf(g.y), a2 = fabsf(g.z), a3 = fabsf(g.w);
    // rank = (#strictly greater) + (#earlier equal)  -> matches lax.top_k ties
    const int r0 = (a1 >  a0) + (a2 >  a0) + (a3 >  a0);
    const int r1 = (a0 >= a1) + (a2 >  a1) + (a3 >  a1);
    const int r2 = (a0 >= a2) + (a1 >= a2) + (a3 >  a2);
    const int r3 = (a0 >= a3) + (a1 >= a3) + (a2 >= a3);
    const bool b0 = r0 < 2, b1 = r1 < 2, b2 = r2 < 2, b3 = r3 < 2;

    const float k0 = b0 ? g.x : 0.0f;
    const float k1 = b1 ? g.y : 0.0f;
    const float k2 = b2 ? g.z : 0.0f;
    const float k3 = b3 ? g.w : 0.0f;

    const int   i0 = b0 ? 0   : (b1 ? 1   : 2);
    const int   i1 = b3 ? 3   : (b2 ? 2   : 1);
    const float p0 = b0 ? g.x : (b1 ? g.y : g.z);
    const float p1 = b3 ? g.w : (b2 ? g.z : g.y);

    codes[tid] = (unsigned int)(i0 | (i1 << 2)) << ((tid & 7) * 4);

    red[0][tid] = g.x + g.y + g.z + g.w;
    red[1][tid] = g.x*g.x + g.y*g.y + g.z*g.z + g.w*g.w;
    red[2][tid] = k0 + k1 + k2 + k3;
    red[3][tid] = k0*k0 + k1*k1 + k2*k2 + k3*k3;
    __syncthreads();

    for (int s = 128; s > 0; s >>= 1) {
        if (tid < s) {
            red[0][tid] += red[0][tid + s];
            red[1][tid] += red[1][tid + s];
            red[2][tid] += red[2][tid + s];
            red[3][tid] += red[3][tid + s];
        }
        __syncthreads();
    }

    if (tid == 0) {
        const float S1 = red[0][0], S2 = red[1][0];
        const float P1 = red[2][0], P2 = red[3][0];
        const float inv_dm1 = 1.0f / (float)(DIM - 1);
        const float var_x  = (S2 - S1 * S1 * (1.0f / DIM)) * inv_dm1;
        float       var_sp = (P2 - P1 * P1 * (1.0f / DIM)) * inv_dm1;
        var_sp = fmaxf(var_sp, VAR_EPS);
        vsh = sqrtf(var_x / var_sp);
    }
    __syncthreads();

    if (tid < IPR) {
        unsigned int c = 0;
        #pragma unroll
        for (int j = 0; j < 8; ++j) c |= codes[tid * 8 + j];
        idxbuf[(size_t)t * IPR + tid] = c;
    }

    const float v = vsh;
    v2h h;
    h.x = (_Float16)(v * p0);
    h.y = (_Float16)(v * p1);
    *reinterpret_cast<v2h*>(apack + (size_t)t * PKD + 2 * tid) = h;
}

// ---------------------------------------------------------------------------
// One-shot W fp32 -> f16 conversion (2 MB result, L2-resident thereafter).
// ---------------------------------------------------------------------------
__global__ __launch_bounds__(256)
void wconv_kernel(const float* __restrict__ W, _Float16* __restrict__ w16)
{
    const size_t i = ((size_t)blockIdx.x * 256 + threadIdx.x) * 4;
    const float4 f = *reinterpret_cast<const float4*>(W + i);
    v4h c = {(_Float16)f.x, (_Float16)f.y, (_Float16)f.z, (_Float16)f.w};
    *reinterpret_cast<v4h*>(w16 + i) = c;
}

// ---------------------------------------------------------------------------
// Pass 2: 2:4 sparse GEMM via V_SWMMAC_F32_16X16X64_F16 with double-buffered
// GLOBAL_LOAD_ASYNC_TO_LDS tile movement (ASYNCcnt pipeline, no VGPR staging).
// Block tile 64x128, 8 waves (2x4), each wave 32x32 = 2x2 accumulators.
// ---------------------------------------------------------------------------
__global__ __launch_bounds__(256)
void spmm_kernel(const _Float16* __restrict__ Ap, const unsigned int* __restrict__ Ix,
                 const _Float16* __restrict__ W16, float* __restrict__ out)
{
    __shared__ _Float16     As[2][BM][BKP];   // packed A tiles: 2 x 4 KB
    __shared__ _Float16     Bs[2][BN][BK];    // f16 B tiles   : 2 x 16 KB
    __shared__ unsigned int Is[2][BM][2];     // 2:4 metadata  : 2 x 512 B

    const int tid  = threadIdx.x;
    const int wave = tid >> 5;
    const int lane = tid & 31;
    const int hi   = lane >> 4;
    const int l16  = lane & 15;

    const int block_m = blockIdx.y * BM;
    const int block_n = blockIdx.x * BN;
    const int wm = (wave >> 2) * 32;
    const int wn = (wave & 3)  * 32;

    // per-thread async-copy slices
    const int a_row = tid >> 2;            // A: 16 B per thread
    const int a_col = (tid & 3) * 8;
    const int b_row = tid >> 1;            // B: 64 B per thread (4 x b128)
    const int b_col = (tid & 1) * 32;

    // issue one K-step's tile movement into buffer `buf` via async-to-LDS.
    // NOTE: INST_OFFSET is added to BOTH the LDS and global addresses, so one
    // base pair + offset:0/16/32/48 walks the 64B B slice on both sides.
    auto issue = [&](int buf, int kk) {
        {
            unsigned long long ga = (unsigned long long)
                (Ap + (size_t)(block_m + a_row) * PKD + kk * BKP + a_col);
            unsigned la = (unsigned)(size_t)&As[buf][a_row][a_col];
            asm volatile("global_load_async_to_lds_b128 %0, %1, off"
                         :: "v"(la), "v"(ga) : "memory");
        }
        {
            unsigned long long gb = (unsigned long long)
                (W16 + (size_t)(block_n + b_row) * DIM + kk * BK + b_col);
            unsigned lb = (unsigned)(size_t)&Bs[buf][b_row][b_col];
            asm volatile("global_load_async_to_lds_b128 %0, %1, off"
                         :: "v"(lb), "v"(gb) : "memory");
            asm volatile("global_load_async_to_lds_b128 %0, %1, off offset:16"
                         :: "v"(lb), "v"(gb) : "memory");
            asm volatile("global_load_async_to_lds_b128 %0, %1, off offset:32"
                         :: "v"(lb), "v"(gb) : "memory");
            asm volatile("global_load_async_to_lds_b128 %0, %1, off offset:48"
                         :: "v"(lb), "v"(gb) : "memory");
        }
        if (tid < BM) {
            unsigned long long gi = (unsigned long long)
                (Ix + (size_t)(block_m + tid) * IPR + kk * 2);
            unsigned li = (unsigned)(size_t)&Is[buf][tid][0];
            asm volatile("global_load_async_to_lds_b64 %0, %1, off"
                         :: "v"(li), "v"(gi) : "memory");
        }
    };

    v8f acc[2][2] = {};

    issue(0, 0);                               // prologue fill

    for (int kk = 0; kk < NK; ++kk) {
        const int cur = kk & 1;

        // tile kk resident: drain this wave's async queue, then sync the block
        asm volatile("s_wait_asynccnt 0x0" ::: "memory");
        __syncthreads();

        // start filling the other buffer while we compute
        if (kk + 1 < NK) issue(cur ^ 1, kk + 1);
        // keep far prefetch warm in GL2 (global_prefetch_b8)
        if (kk + 2 < NK)
            __builtin_prefetch(Ap + (size_t)(block_m + a_row) * PKD
                                   + (kk + 2) * BKP + a_col, 0, 3);

        #pragma unroll
        for (int sm = 0; sm < 2; ++sm) {
            // stored-A layout (16-bit, 16x32 stored): lane row = l16;
            // halves 0..7 -> s = 8*hi.., halves 8..15 -> s = 16+8*hi..
            const _Float16* ap = &As[cur][wm + sm * 16 + l16][0];
            v8h alo = *reinterpret_cast<const v8h*>(ap + 8 * hi);
            v8h ahi = *reinterpret_cast<const v8h*>(ap + 16 + 8 * hi);
            v16h afrag = __builtin_shufflevector(alo, ahi,
                0,1,2,3,4,5,6,7,8,9,10,11,12,13,14,15);

            // index VGPR: 8 group-codes for expanded K [32*hi, 32*hi+32)
            const int aidx = (int)Is[cur][wm + sm * 16 + l16][hi];

            #pragma unroll
            for (int sn = 0; sn < 2; ++sn) {
                // dense B 64x16: lane col = l16; halves 0..15 -> K = 16*hi..,
                // halves 16..31 -> K = 32+16*hi..
                const _Float16* bp = &Bs[cur][wn + sn * 16 + l16][0];
                v16h blo = *reinterpret_cast<const v16h*>(bp + 16 * hi);
                v16h bhi = *reinterpret_cast<const v16h*>(bp + 32 + 16 * hi);
                v32h bfrag = __builtin_shufflevector(blo, bhi,
                    0,1,2,3,4,5,6,7,8,9,10,11,12,13,14,15,
                    16,17,18,19,20,21,22,23,24,25,26,27,28,29,30,31);

                acc[sm][sn] = __builtin_amdgcn_swmmac_f32_16x16x64_f16(
                    false, afrag, false, bfrag, acc[sm][sn], aidx, false, false);
            }
        }
        __syncthreads();   // all waves done reading buf[cur] before it refills
    }

    // ---- epilogue: f32 C/D layout -> lanes 0-15: M=r, lanes 16-31: M=r+8 ----
    #pragma unroll
    for (int sm = 0; sm < 2; ++sm) {
        #pragma unroll
        for (int sn = 0; sn < 2; ++sn) {
            const int col = block_n + wn + sn * 16 + l16;
            #pragma unroll
            for (int r = 0; r < 8; ++r) {
                const int row = block_m + wm + sm * 16 + r + 8 * hi;
                out[(size_t)row * NOUT + col] = acc[sm][sn][r];
            }
        }
    }
}

// ---------------------------------------------------------------------------
// Launch
// ---------------------------------------------------------------------------
extern "C" void kernel_launch(void* const* d_in, const int* in_sizes, int n_in,
                              void* d_out, int out_size, void* d_ws, size_t ws_size,
                              hipStream_t stream)
{
    const float* x = (const float*)d_in[0];   // [4, 8192, 1024] fp32
    const float* w = (const float*)d_in[1];   // [1024, 1024] fp32
    float* out = (float*)d_out;               // [4, 8192, 1024] fp32

    const int T = in_sizes[0] / DIM;          // 32768 rows

    char* ws = (char*)d_ws;
    _Float16*     apack  = (_Float16*)ws;                               // 32 MB
    unsigned int* idxbuf = (unsigned int*)(ws + (size_t)T * PKD * 2);   // +4 MB
    _Float16*     w16    = (_Float16*)(ws + (size_t)T * PKD * 2
                                          + (size_t)T * IPR * 4);       // +2 MB

    wconv_kernel<<<(NOUT * DIM) / 1024, 256, 0, stream>>>(w, w16);
    prep_kernel<<<T, 256, 0, stream>>>(x, apack, idxbuf);

    dim3 grid(NOUT / BN, T / BM);             // (8, 512)
    spmm_kernel<<<grid, 256, 0, stream>>>(apack, idxbuf, w16, out);
}